// ModelNew_50388556317005
// MI455X (gfx1250) — compile-verified
//
#include <hip/hip_runtime.h>

typedef float v2f __attribute__((ext_vector_type(2)));
typedef float v8f __attribute__((ext_vector_type(8)));

// ---- problem constants ----
#define N_BATCH 32
#define CIN     32
#define H_IN    128
#define W_IN    256
#define COUT    64
#define H_OUT   257
#define W_OUT   766
#define CH_STR  (H_OUT * W_OUT)   // 196862, per-output-channel stride

// Zero-fill the 129 even output rows (identically zero: oh even => no valid tap).
__global__ __launch_bounds__(256) void zero_even_rows(float* __restrict__ out) {
    int row = blockIdx.x;                  // 0 .. 32*64*129-1
    int oh  = (row % 129) * 2;             // even oh
    int nc  = row / 129;                   // n*64 + c
    size_t base = ((size_t)nc * H_OUT + oh) * (size_t)W_OUT;
    v2f z = {0.0f, 0.0f};
    v2f* p = (v2f*)(out + base);           // 8B-aligned: 766*4 % 8 == 0
    int i = threadIdx.x;
    if (i < 383)       __builtin_nontemporal_store(z, &p[i]);   // 383 float2 = 766 floats
    if (i + 256 < 383) __builtin_nontemporal_store(z, &p[i + 256]);
}

// One 16(oc) x 16(ow, stride-3 columns) tile via V_WMMA_F32_16X16X4_F32.
// NQ = number of valid width taps for this residue; QA/QB = kernel q indices;
// dw (iw offset) is 1 for the first tap iff NQ==2, else 0.
// xg = x + (n*CIN + g*8)*H_IN*W_IN ; wg = w + (g*8)*16*15   (scalar bases)
template<int NQ, int QA, int QB>
__device__ __forceinline__ v8f tile_mma(const float* __restrict__ xg,
                                        const float* __restrict__ wg,
                                        int m, int kcol, int col, int hi) {
    v8f acc = {0.f, 0.f, 0.f, 0.f, 0.f, 0.f, 0.f, 0.f};
    // scalar height validity per p (ih = m - p, m in 1..128):
    const bool okp[3] = { m <= 127, true, m >= 2 };
    constexpr int S = (24 * NQ) / 4;       // K total = 8*3*NQ; 6 or 12 slices of K=4
#pragma unroll
    for (int s = 0; s < S; ++s) {
        v2f a, b;
#pragma unroll
        for (int j = 0; j < 2; ++j) {
            // Fragment K map (identical for A and B):
            //   lanes 0-15 -> 4s + j ; lanes 16-31 -> 4s + 2 + j
            const int kkL = 4 * s + j;
            const int kkH = kkL + 2;
            const int icL = kkL / (3 * NQ), remL = kkL % (3 * NQ);
            const int icH = kkH / (3 * NQ), remH = kkH % (3 * NQ);
            const int pL = remL / NQ, qiL = remL % NQ;
            const int pH = remH / NQ, qiH = remH % NQ;
            int ic = hi ? icH : icL;
            int p  = hi ? pH  : pL;
            int q  = hi ? (qiH ? QB : QA) : (qiL ? QB : QA);
            int dw = (NQ == 2) ? (hi ? (qiH ? 0 : 1) : (qiL ? 0 : 1)) : 0;

            // A[M=col][K] = w[g*8+ic][col][p][q]   (w shape: 32 x 16 x 3 x 5)
            a[j] = wg[(ic * 16 + col) * 15 + p * 5 + q];

            // B[K][N=col] = x[n][g*8+ic][m-p][kcol+dw], masked.
            // Height mask is scalar (okp[p]); width mask only fails for
            // dw==1 && kcol==255. Clamp index + select => no EXEC divergence.
            int ih = m - p;
            int iw = kcol + dw;
            bool ok = (hi ? okp[pH] : okp[pL]) && (iw < W_IN);
            int xi = (ic * H_IN + (ok ? ih : 0)) * W_IN + (ok ? iw : 0);
            float v = xg[xi];
            b[j] = ok ? v : 0.0f;
        }
        acc = __builtin_amdgcn_wmma_f32_16x16x4_f32(false, a, false, b,
                                                    (short)0, acc, false, false);
    }
    return acc;
}

__global__ __launch_bounds__(256) void deconv_wmma(const float* __restrict__ x,
                                                   const float* __restrict__ w,
                                                   float* __restrict__ out) {
    // 8 waves per block, 1 tile per wave. Tile id layout (innermost first):
    // r(3), t(16), m(128), g(4), n(32) -> 786432 tiles = 98304 blocks.
    // Scalarize the wave-uniform decomposition so it lives in SGPRs and the
    // residue branch is a scalar s_cbranch (keeps EXEC all-1s for WMMA).
    int wslot = __builtin_amdgcn_readfirstlane((int)(threadIdx.x >> 5));
    int wid   = __builtin_amdgcn_readfirstlane((int)blockIdx.x * 8 + wslot);

    int r   = wid % 3;  int tmp = wid / 3;
    int t   = tmp & 15; tmp >>= 4;
    int m   = 1 + (tmp & 127); tmp >>= 7;   // oh = 2m-1, m in 1..128
    int g   = tmp & 3;
    int n   = tmp >> 2;

    int lane = threadIdx.x & 31;
    int col  = lane & 15;      // N column of tile (also M row for A fragment)
    int hi   = lane >> 4;      // upper lane half
    int kcol = t * 16 + col;   // iw base index; ow = r + 3*kcol

    const float* xg = x + (size_t)(n * CIN + g * 8) * (H_IN * W_IN);
    const float* wg = w + (g * 8) * (16 * 15);

    v8f acc;
    if (r == 0)      acc = tile_mma<1, 2, 2>(xg, wg, m, kcol, col, hi);
    else if (r == 1) acc = tile_mma<2, 0, 3>(xg, wg, m, kcol, col, hi);
    else             acc = tile_mma<2, 1, 4>(xg, wg, m, kcol, col, hi);

    int oh = 2 * m - 1;
    int ow = r + 3 * kcol;
    if (ow < W_OUT) {
        // D layout: VGPR j, lanes 0-15 -> M=j ; lanes 16-31 -> M=j+8 ; N=lane&15
        size_t base = (((size_t)(n * COUT + g * 16 + 8 * hi) * H_OUT + oh)
                       * (size_t)W_OUT) + ow;
#pragma unroll
        for (int j = 0; j < 8; ++j)
            __builtin_nontemporal_store(acc[j], &out[base + (size_t)j * CH_STR]);
    }
}

extern "C" void kernel_launch(void* const* d_in, const int* in_sizes, int n_in,
                              void* d_out, int out_size, void* d_ws, size_t ws_size,
                              hipStream_t stream) {
    const float* x = (const float*)d_in[0];
    const float* w = (const float*)d_in[1];
    float* out = (float*)d_out;

    // 1) zero the 129 even rows: 32*64*129 = 264192 blocks
    zero_even_rows<<<264192, 256, 0, stream>>>(out);

    // 2) WMMA compute of the 128 odd rows: 786432 tiles / 8 waves = 98304 blocks
    deconv_wmma<<<98304, 256, 0, stream>>>(x, w, out);
}